// BiggerMLP_34961033790088
// MI455X (gfx1250) — compile-verified
//
#include <hip/hip_runtime.h>

typedef __attribute__((ext_vector_type(2))) float v2f;
typedef __attribute__((ext_vector_type(8))) float v8f;

#define LDH 68   // activation LDS row stride (floats): 68 % 64 == 4 -> conflict-free

__device__ __forceinline__ v8f wmma4(v2f a, v2f b, v8f c) {
    // D = A(16x4) * B(4x16) + C(16x16), fp32
    return __builtin_amdgcn_wmma_f32_16x16x4_f32(
        false, a, false, b, (short)0, c, false, false);
}

__global__ __launch_bounds__(256) void mlp5_wmma_kernel(
    const float* __restrict__ x,
    const float* __restrict__ w1, const float* __restrict__ b1,
    const float* __restrict__ w2, const float* __restrict__ b2,
    const float* __restrict__ w3, const float* __restrict__ b3,
    const float* __restrict__ w4, const float* __restrict__ b4,
    const float* __restrict__ w5, const float* __restrict__ b5,
    float* __restrict__ out, int nrows)
{
    // Per-wave activation staging (16 rows x up to 64 cols, stride 68)
    __shared__ __align__(16) float s_h[8][16 * LDH];
    // Lane-major B-fragment storage: frag f of lane L at [f][L][0..1]
    // b64 load hits banks {2L, 2L+1} -> all 64 banks exactly once, conflict-free.
    __shared__ __align__(16) float s_B2[32][32][2];   // layer2: 16 ksteps x 2 ntiles
    __shared__ __align__(16) float s_B3[8][32][2];    // layer3: 8 ksteps
    __shared__ __align__(16) float s_B4[4][32][2];    // layer4: 4 ksteps (N pad 8->16)
    __shared__ __align__(16) float s_B5[2][32][2];    // layer5: 2 ksteps (N pad 2->16)

    const int tid  = threadIdx.x;
    const int lane = tid & 31;
    const int wid  = tid >> 5;
    const int nlo  = lane & 15;                 // col (B/C) or row (A) within tile
    const int khi  = (lane & 16) ? 2 : 0;       // K sub-offset for A/B fragments
    const int rhi  = (lane & 16) ? 8 : 0;       // M offset for C/D tiles
    float* hw = &s_h[wid][0];

    // ---- wave 0 builds B2..B5 fragments straight from global into LDS ----
    if (wid == 0) {
#pragma unroll
        for (int k = 0; k < 16; ++k)
#pragma unroll
            for (int nt = 0; nt < 2; ++nt) {
                int kk = 4 * k + khi, col = 16 * nt + nlo;   // w2 is (32,64): w2[n][k]
                s_B2[k * 2 + nt][lane][0] = w2[col * 64 + kk];
                s_B2[k * 2 + nt][lane][1] = w2[col * 64 + kk + 1];
            }
#pragma unroll
        for (int k = 0; k < 8; ++k) {
            int kk = 4 * k + khi;                            // w3 is (16,32)
            s_B3[k][lane][0] = w3[nlo * 32 + kk];
            s_B3[k][lane][1] = w3[nlo * 32 + kk + 1];
        }
#pragma unroll
        for (int k = 0; k < 4; ++k) {
            int kk = 4 * k + khi;                            // w4 is (8,16), N padded
            s_B4[k][lane][0] = (nlo < 8) ? w4[nlo * 16 + kk] : 0.f;
            s_B4[k][lane][1] = (nlo < 8) ? w4[nlo * 16 + kk + 1] : 0.f;
        }
#pragma unroll
        for (int k = 0; k < 2; ++k) {
            int kk = 4 * k + khi;                            // w5 is (2,8), N padded
            s_B5[k][lane][0] = (nlo < 2) ? w5[nlo * 8 + kk] : 0.f;
            s_B5[k][lane][1] = (nlo < 2) ? w5[nlo * 8 + kk + 1] : 0.f;
        }
    }

    // ---- B1 stays in registers (hottest layer), K padded 30 -> 32 ----
    v2f B1[8][4];
#pragma unroll
    for (int k = 0; k < 8; ++k)
#pragma unroll
        for (int nt = 0; nt < 4; ++nt) {
            int kk = 4 * k + khi, col = 16 * nt + nlo;       // w1 is (64,30)
            B1[k][nt].x = (kk < 30) ? w1[col * 30 + kk] : 0.f;
            B1[k][nt].y = (kk + 1 < 30) ? w1[col * 30 + kk + 1] : 0.f;
        }

    // biases, per-lane (column = lane-derived)
    float bv1[4], bv2[2];
#pragma unroll
    for (int nt = 0; nt < 4; ++nt) bv1[nt] = b1[16 * nt + nlo];
#pragma unroll
    for (int nt = 0; nt < 2; ++nt) bv2[nt] = b2[16 * nt + nlo];
    const float bv3 = b3[nlo];
    const float bv4 = (nlo < 8) ? b4[nlo] : 0.f;
    const float bv5 = (nlo < 2) ? b5[nlo] : 0.f;

    __syncthreads();

    auto splat8 = [](float v) { v8f r;
#pragma unroll
        for (int i = 0; i < 8; ++i) r[i] = v; return r; };
    auto relu8 = [](v8f c) {
#pragma unroll
        for (int i = 0; i < 8; ++i) c[i] = fmaxf(c[i], 0.f); return c; };
    auto add8 = [](v8f a, v8f b) { v8f r;
#pragma unroll
        for (int i = 0; i < 8; ++i) r[i] = a[i] + b[i]; return r; };
    auto afrag = [&](int k0) -> v2f {                     // A fragment from staged h
        return *(const v2f*)(hw + nlo * LDH + k0 + khi);
    };
    auto ldB = [&](const float* base, int f) -> v2f {     // lane-major B fragment
        return *(const v2f*)(base + (size_t)f * 64 + lane * 2);
    };
    auto store_tile = [&](v8f c, int nb) {
#pragma unroll
        for (int r = 0; r < 8; ++r) hw[(rhi + r) * LDH + nb + nlo] = c[r];
    };

    const float* pB2 = &s_B2[0][0][0];
    const float* pB3 = &s_B3[0][0][0];
    const float* pB4 = &s_B4[0][0][0];
    const float* pB5 = &s_B5[0][0][0];

    const int ntiles = (nrows + 15) >> 4;
    const int gwave  = (blockIdx.x << 3) + wid;
    const int nwaves = gridDim.x << 3;

    for (int t = gwave; t < ntiles; t += nwaves) {
        const int row0 = t << 4;
        int rr = row0 + nlo; if (rr >= nrows) rr = nrows - 1;    // tail clamp
        const float* xrow = x + (size_t)rr * 30;

        // A fragments of x, K padded 30 -> 32
        v2f ax[8];
#pragma unroll
        for (int k = 0; k < 7; ++k) ax[k] = *(const v2f*)(xrow + 4 * k + khi);
        {
            v2f tl = *(const v2f*)(xrow + 28);                   // cols 28,29
            ax[7].x = (khi == 0) ? tl.x : 0.f;                   // cols 30,31 pad
            ax[7].y = (khi == 0) ? tl.y : 0.f;
        }

        // ---- layer 1: 16x30 @ 30x64, 4 independent N-chains ----
        v8f c0 = splat8(bv1[0]), c1 = splat8(bv1[1]), c2 = splat8(bv1[2]), c3 = splat8(bv1[3]);
#pragma unroll
        for (int k = 0; k < 8; ++k) {
            c0 = wmma4(ax[k], B1[k][0], c0);
            c1 = wmma4(ax[k], B1[k][1], c1);
            c2 = wmma4(ax[k], B1[k][2], c2);
            c3 = wmma4(ax[k], B1[k][3], c3);
        }
        store_tile(relu8(c0), 0);  store_tile(relu8(c1), 16);
        store_tile(relu8(c2), 32); store_tile(relu8(c3), 48);

        // ---- layer 2: 16x64 @ 64x32, 4 independent chains (2 N x 2 K-halves) ----
        v8f d0a = splat8(bv2[0]), d1a = splat8(bv2[1]);
        v8f d0b = splat8(0.f),    d1b = splat8(0.f);
#pragma unroll
        for (int k = 0; k < 8; ++k) {
            v2f a1 = afrag(4 * k);
            v2f a2 = afrag(4 * (k + 8));
            d0a = wmma4(a1, ldB(pB2, 2 * k + 0), d0a);
            d1a = wmma4(a1, ldB(pB2, 2 * k + 1), d1a);
            d0b = wmma4(a2, ldB(pB2, 2 * (k + 8) + 0), d0b);
            d1b = wmma4(a2, ldB(pB2, 2 * (k + 8) + 1), d1b);
        }
        store_tile(relu8(add8(d0a, d0b)), 0);
        store_tile(relu8(add8(d1a, d1b)), 16);

        // ---- layer 3: 16x32 @ 32x16, 2 K-split chains ----
        v8f e0 = splat8(bv3), e1 = splat8(0.f);
#pragma unroll
        for (int k = 0; k < 4; ++k) {
            e0 = wmma4(afrag(4 * k),       ldB(pB3, k),     e0);
            e1 = wmma4(afrag(4 * (k + 4)), ldB(pB3, k + 4), e1);
        }
        store_tile(relu8(add8(e0, e1)), 0);

        // ---- layer 4: 16x16 @ 16x8 (N pad 16), 2 K-split chains ----
        v8f f0 = splat8(bv4), f1 = splat8(0.f);
        f0 = wmma4(afrag(0),  ldB(pB4, 0), f0);
        f1 = wmma4(afrag(8),  ldB(pB4, 2), f1);
        f0 = wmma4(afrag(4),  ldB(pB4, 1), f0);
        f1 = wmma4(afrag(12), ldB(pB4, 3), f1);
        store_tile(relu8(add8(f0, f1)), 0);

        // ---- layer 5: 16x8 @ 8x2 (N pad 16), no ReLU ----
        v8f g0 = splat8(bv5), g1 = splat8(0.f);
        g0 = wmma4(afrag(0), ldB(pB5, 0), g0);
        g1 = wmma4(afrag(4), ldB(pB5, 1), g1);
        store_tile(add8(g0, g1), 0);

        // coalesced output: lane -> out[row0*2 + lane]
        float val = hw[(lane >> 1) * LDH + (lane & 1)];
        size_t oi = (size_t)row0 * 2 + lane;
        if (oi < (size_t)nrows * 2) out[oi] = val;
    }
}

extern "C" void kernel_launch(void* const* d_in, const int* in_sizes, int n_in,
                              void* d_out, int out_size, void* d_ws, size_t ws_size,
                              hipStream_t stream) {
    (void)n_in; (void)d_ws; (void)ws_size; (void)out_size;
    const float* x  = (const float*)d_in[0];
    const float* w1 = (const float*)d_in[1];  const float* b1 = (const float*)d_in[2];
    const float* w2 = (const float*)d_in[3];  const float* b2 = (const float*)d_in[4];
    const float* w3 = (const float*)d_in[5];  const float* b3 = (const float*)d_in[6];
    const float* w4 = (const float*)d_in[7];  const float* b4 = (const float*)d_in[8];
    const float* w5 = (const float*)d_in[9];  const float* b5 = (const float*)d_in[10];
    float* out = (float*)d_out;

    int nrows  = in_sizes[0] / 30;
    int ntiles = (nrows + 15) / 16;
    int blocks = (ntiles + 7) / 8;
    if (blocks > 2048) blocks = 2048;
    if (blocks < 1) blocks = 1;

    mlp5_wmma_kernel<<<blocks, 256, 0, stream>>>(
        x, w1, b1, w2, b2, w3, b3, w4, b4, w5, b5, out, nrows);
}